// MultiHeadAttention_47931835023953
// MI455X (gfx1250) — compile-verified
//
#include <hip/hip_runtime.h>
#include <math.h>

// Problem constants (from reference)
#define N_NODES 50000
#define E_EDGES 800000
#define D_IN    128          // == H*DH
#define NHEAD   4
#define DHEAD   32
#define NEG_SLOPE 0.01f
#define EPS_F   1e-16f

typedef __attribute__((ext_vector_type(2))) float v2f;
typedef __attribute__((ext_vector_type(8))) float v8f;

// ---------------------------------------------------------------------------
// Kernel 1: fused dual GEMM via V_WMMA_F32_16X16X4_F32
//   h   = x @ W_proj   (N x 128)
//   out = x @ W_skip   (N x 128)   <- skip connection pre-seeded into d_out
// Block = 256 threads = 8 waves; wave w owns column tile w (8 tiles x 16 = 128).
// grid.x = N/16 = 3125 row tiles (exact).
// ---------------------------------------------------------------------------
__global__ __launch_bounds__(256) void gat_gemm_kernel(
    const float* __restrict__ x,
    const float* __restrict__ Wp,
    const float* __restrict__ Ws,
    float* __restrict__ h,
    float* __restrict__ out)
{
    const int wave = threadIdx.x >> 5;   // 0..7 -> column tile
    const int lane = threadIdx.x & 31;
    const int half = lane >> 4;          // 0: K pair {0,1}; 1: K pair {2,3}
    const int lr   = lane & 15;
    const int rowBase = blockIdx.x << 4;
    const int colBase = wave << 4;
    const int row = rowBase + lr;        // A row for this lane
    const int col = colBase + lr;        // B/C column for this lane

    v8f accP = {};
    v8f accS = {};

    for (int kb = 0; kb < D_IN; kb += 4) {
        const int ka = kb + 2 * half;
        // A 16x4 f32: lane holds x[row, ka], x[row, ka+1] (8B aligned: ka even)
        const float2 av = *(const float2*)(x + (size_t)row * D_IN + ka);
        v2f a; a.x = av.x; a.y = av.y;
        // B 4x16 f32 (row-major K x N): mirrored layout
        v2f bp, bs;
        bp.x = Wp[(ka + 0) * D_IN + col];
        bp.y = Wp[(ka + 1) * D_IN + col];
        bs.x = Ws[(ka + 0) * D_IN + col];
        bs.y = Ws[(ka + 1) * D_IN + col];
        accP = __builtin_amdgcn_wmma_f32_16x16x4_f32(
            false, a, false, bp, (short)0, accP, false, false);
        accS = __builtin_amdgcn_wmma_f32_16x16x4_f32(
            false, a, false, bs, (short)0, accS, false, false);
    }

    // C/D layout: VGPR r holds M = r + 8*half, N = col
#pragma unroll
    for (int r = 0; r < 8; ++r) {
        const int m = rowBase + 8 * half + r;
        h[(size_t)m * D_IN + col]   = accP[r];
        out[(size_t)m * D_IN + col] = accS[r];
    }
}

// ---------------------------------------------------------------------------
// Kernel 2: init segment-max to -inf, denom to 0 (N*H each)
// ---------------------------------------------------------------------------
__global__ void gat_init_kernel(float* __restrict__ emax,
                                float* __restrict__ denom, int n)
{
    int i = blockIdx.x * blockDim.x + threadIdx.x;
    if (i < n) { emax[i] = -INFINITY; denom[i] = 0.0f; }
}

// ---------------------------------------------------------------------------
// Kernel 3: per-node attention scores  s_src[n,h] = <h[n,h,:], a_src[h]>,
//                                      s_dst[n,h] = <h[n,h,:], a_dst[h]>
// This factorization cuts the per-edge gather from 1KB to 32B.
// ---------------------------------------------------------------------------
__global__ void gat_scores_kernel(const float* __restrict__ h,
                                  const float* __restrict__ att,  // (H, 2*DH)
                                  float* __restrict__ ssrc,
                                  float* __restrict__ sdst)
{
    int idx = blockIdx.x * blockDim.x + threadIdx.x;  // n*NHEAD + hh
    if (idx >= N_NODES * NHEAD) return;
    const int n  = idx >> 2;
    const int hh = idx & 3;
    const float* hp = h + (size_t)n * D_IN + hh * DHEAD;
    const float* as = att + hh * 2 * DHEAD;
    const float* ad = as + DHEAD;
    float ss = 0.0f, sd = 0.0f;
#pragma unroll
    for (int d = 0; d < DHEAD; d += 4) {
        float4 v  = *(const float4*)(hp + d);
        float4 a1 = *(const float4*)(as + d);
        float4 a2 = *(const float4*)(ad + d);
        ss += v.x * a1.x + v.y * a1.y + v.z * a1.z + v.w * a1.w;
        sd += v.x * a2.x + v.y * a2.y + v.z * a2.z + v.w * a2.w;
    }
    ssrc[idx] = ss;
    sdst[idx] = sd;
}

// ---------------------------------------------------------------------------
// Kernel 4: edge logits (leaky relu) + atomic segment max over dst
// one thread per (edge, head)
// ---------------------------------------------------------------------------
__global__ void gat_logits_kernel(const int* __restrict__ ei,
                                  const float* __restrict__ ssrc,
                                  const float* __restrict__ sdst,
                                  float* __restrict__ ebuf,
                                  float* __restrict__ emax)
{
    int idx = blockIdx.x * blockDim.x + threadIdx.x;  // e*NHEAD + hh
    if (idx >= E_EDGES * NHEAD) return;
    const int e  = idx >> 2;
    const int hh = idx & 3;
    const int s  = ei[e];
    const int d  = ei[E_EDGES + e];
    float v = ssrc[s * NHEAD + hh] + sdst[d * NHEAD + hh];
    v = (v > 0.0f) ? v : NEG_SLOPE * v;
    ebuf[idx] = v;
    atomicMax(emax + d * NHEAD + hh, v);   // global_atomic_max_num_f32
}

// ---------------------------------------------------------------------------
// Kernel 5: e_exp = exp(e - emax[dst]) in place; atomic segment sum -> denom
// ---------------------------------------------------------------------------
__global__ void gat_expsum_kernel(const int* __restrict__ ei,
                                  float* __restrict__ ebuf,
                                  const float* __restrict__ emax,
                                  float* __restrict__ denom)
{
    int idx = blockIdx.x * blockDim.x + threadIdx.x;
    if (idx >= E_EDGES * NHEAD) return;
    const int e  = idx >> 2;
    const int hh = idx & 3;
    const int d  = ei[E_EDGES + e];
    float ex = expf(ebuf[idx] - emax[d * NHEAD + hh]);
    ebuf[idx] = ex;
    atomicAdd(denom + d * NHEAD + hh, ex);
}

// ---------------------------------------------------------------------------
// Kernel 6: out[dst, f] += (e_exp / (denom[dst]+eps)) * h[src, f]
// one thread per (edge, feature); a wave covers exactly one (edge, head),
// so ebuf/denom loads are wave-uniform.
// ---------------------------------------------------------------------------
__global__ void gat_aggregate_kernel(const int* __restrict__ ei,
                                     const float* __restrict__ ebuf,
                                     const float* __restrict__ denom,
                                     const float* __restrict__ h,
                                     float* __restrict__ out)
{
    const int tid = blockIdx.x * blockDim.x + threadIdx.x;  // < E*128 (exact grid)
    const int e  = tid >> 7;
    const int f  = tid & 127;
    const int hh = f >> 5;
    const int s  = ei[e];
    const int d  = ei[E_EDGES + e];
    const float alpha = ebuf[e * NHEAD + hh] / (denom[d * NHEAD + hh] + EPS_F);
    atomicAdd(out + (size_t)d * D_IN + f, alpha * h[(size_t)s * D_IN + f]);
}

// ---------------------------------------------------------------------------
extern "C" void kernel_launch(void* const* d_in, const int* in_sizes, int n_in,
                              void* d_out, int out_size, void* d_ws, size_t ws_size,
                              hipStream_t stream)
{
    const float* x   = (const float*)d_in[0];   // (N, 128)
    const float* Wp  = (const float*)d_in[1];   // (128, 128)
    const float* att = (const float*)d_in[2];   // (4, 64)
    const float* Ws  = (const float*)d_in[3];   // (128, 128)
    const int*   ei  = (const int*)d_in[4];     // (2, E)
    float* out = (float*)d_out;                 // (N, 128)

    // workspace layout (f32): h | ssrc | sdst | ebuf | emax | denom  (~41.6 MB)
    float* h     = (float*)d_ws;
    float* ssrc  = h    + (size_t)N_NODES * D_IN;
    float* sdst  = ssrc + (size_t)N_NODES * NHEAD;
    float* ebuf  = sdst + (size_t)N_NODES * NHEAD;
    float* emax  = ebuf + (size_t)E_EDGES * NHEAD;
    float* denom = emax + (size_t)N_NODES * NHEAD;

    // 1. dual GEMM (h -> ws, skip -> out)
    gat_gemm_kernel<<<N_NODES / 16, 256, 0, stream>>>(x, Wp, Ws, h, out);

    // 2. init max/denom
    const int nh = N_NODES * NHEAD;
    gat_init_kernel<<<(nh + 255) / 256, 256, 0, stream>>>(emax, denom, nh);

    // 3. per-node scores
    gat_scores_kernel<<<(nh + 255) / 256, 256, 0, stream>>>(h, att, ssrc, sdst);

    // 4. edge logits + segment max
    const int eh = E_EDGES * NHEAD;
    gat_logits_kernel<<<(eh + 255) / 256, 256, 0, stream>>>(ei, ssrc, sdst, ebuf, emax);

    // 5. exp + segment sum
    gat_expsum_kernel<<<(eh + 255) / 256, 256, 0, stream>>>(ei, ebuf, emax, denom);

    // 6. weighted aggregation into out (on top of skip)
    gat_aggregate_kernel<<<(E_EDGES * D_IN) / 256, 256, 0, stream>>>(ei, ebuf, denom, h, out);
}